// TensorSensing_14250701488848
// MI455X (gfx1250) — compile-verified
//
#include <hip/hip_runtime.h>
#include <hip/hip_bf16.h>

// CDNA5 / gfx1250 Tucker encode->relu->decode pipeline.
// Big mode contractions: double-buffered, async-LDS-staged batched GEMM on
// V_WMMA_F32_16X16X4_F32:
//   - workgroup (128 threads = 4 waves) computes a 128x64 C tile
//   - per K-block of 32: A(128x32) and B(32x64) copied global->LDS with
//     GLOBAL_LOAD_ASYNC_TO_LDS_B32/B128 (ASYNCcnt), double-buffered so the
//     next block's copy overlaps the current block's 64 WMMAs
//   - each wave: 32(M) x 64(N) strip = 8 f32 accumulators (8 WMMA / K-step)
// Tiny channel contractions (C=3 <-> d=2): elementwise VALU kernels.

typedef __attribute__((ext_vector_type(2))) float v2f;
typedef __attribute__((ext_vector_type(8))) float v8f;

#define TM 128   // workgroup M tile
#define TN 64    // workgroup N tile
#define TK 32    // K block staged in LDS
#define LDA 33   // sA row stride (pad: conflict-free row-major reads)
#define LDB 72   // sB row stride (pad: half-wave K-split reads hit disjoint banks)

// Strided batched GEMM:  C[z][m][n] = sum_k A[z][m][k] * B[z][k][n]
//   A elem: A + z1*a_z1 + z0*a_z0 + m*a_ms + k*a_ks
//   B elem: B + z1*b_z1 + z0*b_z0 + k*b_ks + n          (b_ns == 1 always here)
//   C elem: C + z1*c_z1 + z0*c_z0 + m*c_ms + n*c_ns
// blockIdx = (n_tile, m_tile, z);  z1 = z / z0_count, z0 = z % z0_count.
struct GemmDesc {
  const float* A;
  const float* B;
  float*       C;
  int K;
  int z0_count;
  long a_z0, a_z1, b_z0, b_z1, c_z0, c_z1;
  int a_ms, a_ks, b_ks, c_ms, c_ns;
};

#define WMMA_F32(acc, a, b) \
  acc = __builtin_amdgcn_wmma_f32_16x16x4_f32(false, (a), false, (b), (short)0, (acc), false, false)

// Async global->LDS copies (CDNA5 Tensor/async path, tracked by ASYNCcnt).
__device__ __forceinline__ void async_copy_b32(unsigned lds_off, const float* gptr) {
  asm volatile("global_load_async_to_lds_b32 %0, %1, off"
               :: "v"(lds_off), "v"((unsigned long long)gptr) : "memory");
}
__device__ __forceinline__ void async_copy_b128(unsigned lds_off, const float* gptr) {
  asm volatile("global_load_async_to_lds_b128 %0, %1, off"
               :: "v"(lds_off), "v"((unsigned long long)gptr) : "memory");
}
__device__ __forceinline__ void wait_async0() {
  asm volatile("s_wait_asynccnt 0x0" ::: "memory");
}

__global__ __launch_bounds__(128) void wmma_gemm_f32(GemmDesc g) {
  __shared__ float sA[2][TM * LDA];  // 2 x 16,896 B
  __shared__ float sB[2][TK * LDB];  // 2 x  9,216 B

  const int tid  = threadIdx.x;
  const int lane = tid & 31;
  const int wid  = tid >> 5;

  const int z  = blockIdx.z;
  const int z1 = z / g.z0_count;
  const int z0 = z % g.z0_count;

  const float* Ab = g.A + (long)z1 * g.a_z1 + (long)z0 * g.a_z0;
  const float* Bb = g.B + (long)z1 * g.b_z1 + (long)z0 * g.b_z0;
  float*       Cb = g.C + (long)z1 * g.c_z1 + (long)z0 * g.c_z0;

  const int m0 = blockIdx.y * TM;
  const int n0 = blockIdx.x * TN;

  // --- staging assignments (per thread) ---
  // A: thread t streams global row (m0+t) of A into sA row t (async b32 gather)
  const float* a_row_p = Ab + (long)(m0 + tid) * g.a_ms;
  const unsigned aoff0 = (unsigned)(unsigned long long)&sA[0][tid * LDA];
  const unsigned aoff1 = (unsigned)(unsigned long long)&sA[1][tid * LDA];
  // B: thread t copies 16 contiguous floats of row (t>>2), col base (t&3)*16
  const int bk = tid >> 2;
  const int bn = (tid & 3) * 16;
  const float* b_row_p = Bb + (long)bk * g.b_ks + n0 + bn;
  const unsigned boff0 = (unsigned)(unsigned long long)&sB[0][bk * LDB + bn];
  const unsigned boff1 = (unsigned)(unsigned long long)&sB[1][bk * LDB + bn];

  // --- compute assignments (per lane) ---
  const int half = lane >> 4;       // K sub-offset select (ISA 16x4 f32 A layout)
  const int l16  = lane & 15;
  const int mr0  = wid * 32 + l16;  // local A row, first 16-row M tile
  const int mr1  = mr0 + 16;        // second M tile

  v8f acc00 = {}, acc01 = {}, acc02 = {}, acc03 = {};
  v8f acc10 = {}, acc11 = {}, acc12 = {}, acc13 = {};

  // issue async stage of one K-block into buffer `buf`
  auto stage = [&](int buf, int k0) {
    const unsigned aoff = buf ? aoff1 : aoff0;
    const float* ap = a_row_p + (long)k0 * g.a_ks;
#pragma unroll
    for (int kk = 0; kk < TK; ++kk)
      async_copy_b32(aoff + 4u * kk, ap + (long)kk * g.a_ks);

    const unsigned boff = buf ? boff1 : boff0;
    const float* bp = b_row_p + (long)k0 * g.b_ks;
#pragma unroll
    for (int q = 0; q < 4; ++q)
      async_copy_b128(boff + 16u * q, bp + 4 * q);
  };

  stage(0, 0);  // prologue: stage first K-block

  int cur = 0;
  for (int k0 = 0; k0 < g.K; k0 += TK) {
    wait_async0();    // my async copies for buf[cur] (and older) complete
    __syncthreads();  // everyone's copies done; everyone done reading buf[cur^1]

    if (k0 + TK < g.K) stage(cur ^ 1, k0 + TK);  // overlap next copy w/ compute

    const float* cA = sA[cur];
    const float* cB = sB[cur];

    // ---- compute: 8 K-steps x 8 WMMA ----
#pragma unroll
    for (int kk = 0; kk < TK; kk += 4) {
      const int ka = kk + 2 * half;  // this lane-half's K pair

      v2f a0, a1;
      a0.x = cA[mr0 * LDA + ka]; a0.y = cA[mr0 * LDA + ka + 1];
      a1.x = cA[mr1 * LDA + ka]; a1.y = cA[mr1 * LDA + ka + 1];

      v2f b0, b1, b2, b3;
      b0.x = cB[ka * LDB + l16 +  0]; b0.y = cB[(ka + 1) * LDB + l16 +  0];
      b1.x = cB[ka * LDB + l16 + 16]; b1.y = cB[(ka + 1) * LDB + l16 + 16];
      b2.x = cB[ka * LDB + l16 + 32]; b2.y = cB[(ka + 1) * LDB + l16 + 32];
      b3.x = cB[ka * LDB + l16 + 48]; b3.y = cB[(ka + 1) * LDB + l16 + 48];

      WMMA_F32(acc00, a0, b0);
      WMMA_F32(acc01, a0, b1);
      WMMA_F32(acc02, a0, b2);
      WMMA_F32(acc03, a0, b3);
      WMMA_F32(acc10, a1, b0);
      WMMA_F32(acc11, a1, b1);
      WMMA_F32(acc12, a1, b2);
      WMMA_F32(acc13, a1, b3);
    }
    cur ^= 1;
  }

  // ---- store: C/D 16x16 f32 layout: lane(l16)=col, VGPR i = row half*8+i ----
  const int cn  = n0 + l16;
  const int cm0 = m0 + wid * 32 + (half << 3);       // first M tile rows
  const int cm1 = cm0 + 16;                          // second M tile rows
#pragma unroll
  for (int i = 0; i < 8; ++i) {
    const long r0 = (long)(cm0 + i) * g.c_ms;
    Cb[r0 + (long)(cn +  0) * g.c_ns] = acc00[i];
    Cb[r0 + (long)(cn + 16) * g.c_ns] = acc01[i];
    Cb[r0 + (long)(cn + 32) * g.c_ns] = acc02[i];
    Cb[r0 + (long)(cn + 48) * g.c_ns] = acc03[i];
    const long r1 = (long)(cm1 + i) * g.c_ms;
    Cb[r1 + (long)(cn +  0) * g.c_ns] = acc10[i];
    Cb[r1 + (long)(cn + 16) * g.c_ns] = acc11[i];
    Cb[r1 + (long)(cn + 32) * g.c_ns] = acc12[i];
    Cb[r1 + (long)(cn + 48) * g.c_ns] = acc13[i];
  }
}

// Encode mode-3 (C=3 -> d=2) + ReLU. One thread per (b,h,w) pixel.
__global__ __launch_bounds__(256) void encode_mode3_relu(
    const float* __restrict__ Y2, const float* __restrict__ P3e,
    float* __restrict__ Y3, int npix) {
  int i = blockIdx.x * blockDim.x + threadIdx.x;
  if (i >= npix) return;
  float c0 = Y2[3 * i + 0], c1 = Y2[3 * i + 1], c2 = Y2[3 * i + 2];
  float d0 = c0 * P3e[0] + c1 * P3e[2] + c2 * P3e[4];  // P3e [3,2]
  float d1 = c0 * P3e[1] + c1 * P3e[3] + c2 * P3e[5];
  Y3[2 * i + 0] = fmaxf(d0, 0.0f);
  Y3[2 * i + 1] = fmaxf(d1, 0.0f);
}

// Decode mode-3 (d=2 -> C=3). One thread per (b,H,W) pixel.
__global__ __launch_bounds__(256) void decode_mode3(
    const float* __restrict__ Z2, const float* __restrict__ P3d,
    float* __restrict__ out, int npix) {
  int i = blockIdx.x * blockDim.x + threadIdx.x;
  if (i >= npix) return;
  float d0 = Z2[2 * i + 0], d1 = Z2[2 * i + 1];
  out[3 * i + 0] = d0 * P3d[0] + d1 * P3d[3];          // P3d [2,3]
  out[3 * i + 1] = d0 * P3d[1] + d1 * P3d[4];
  out[3 * i + 2] = d0 * P3d[2] + d1 * P3d[5];
}

extern "C" void kernel_launch(void* const* d_in, const int* in_sizes, int n_in,
                              void* d_out, int out_size, void* d_ws, size_t ws_size,
                              hipStream_t stream) {
  (void)in_sizes; (void)n_in; (void)out_size; (void)ws_size;

  const float* x   = (const float*)d_in[0];  // [128,256,256,3]
  const float* P1e = (const float*)d_in[1];  // [256,128]
  const float* P2e = (const float*)d_in[2];  // [256,128]
  const float* P3e = (const float*)d_in[3];  // [3,2]
  const float* P1d = (const float*)d_in[4];  // [128,256]
  const float* P2d = (const float*)d_in[5];  // [128,256]
  const float* P3d = (const float*)d_in[6];  // [2,3]
  float* out = (float*)d_out;                // [128,256,256,3]

  // Workspace layout (floats), liveness-based reuse. Peak ~112 MiB.
  float* w  = (float*)d_ws;
  float* Y1 = w + 0;         // [128,128,256,3] = 12,582,912 f
  float* Y2 = w + 12582912;  // [128,128,128,3] =  6,291,456 f
  float* Y3 = w + 0;         // [128,128,128,2] =  4,194,304 f (reuses Y1)
  float* Z1 = w + 4194304;   // [128,256,128,2] =  8,388,608 f
  float* Z2 = w + 12582912;  // [128,256,256,2] = 16,777,216 f (reuses Y2)

  // ---- Encode mode-1: Y1[b,k,wc] = sum_h P1e[h,k] * x[b,h,wc] ----
  {
    GemmDesc g;
    g.A = P1e; g.B = x; g.C = Y1;
    g.K = 256; g.z0_count = 1;
    g.a_z0 = 0; g.a_z1 = 0;
    g.b_z0 = 0; g.b_z1 = 256L * 768;
    g.c_z0 = 0; g.c_z1 = 128L * 768;
    g.a_ms = 1;   g.a_ks = 128;      // A[m][k] = P1e[k,m]
    g.b_ks = 768;
    g.c_ms = 768; g.c_ns = 1;
    wmma_gemm_f32<<<dim3(768 / TN, 128 / TM, 128), 128, 0, stream>>>(g);
  }

  // ---- Encode mode-2: Y2[b,h,k,c] = sum_w Y1[b,h,w,c] * P2e[w,k] ----
  {
    GemmDesc g;
    g.A = Y1; g.B = P2e; g.C = Y2;
    g.K = 256; g.z0_count = 3;       // z1 = b, z0 = c
    g.a_z0 = 1; g.a_z1 = 128L * 256 * 3;
    g.b_z0 = 0; g.b_z1 = 0;
    g.c_z0 = 1; g.c_z1 = 128L * 128 * 3;
    g.a_ms = 768; g.a_ks = 3;        // Y1[b,m,w,c]
    g.b_ks = 128;
    g.c_ms = 384; g.c_ns = 3;        // Y2[b,m,n,c]
    wmma_gemm_f32<<<dim3(128 / TN, 128 / TM, 128 * 3), 128, 0, stream>>>(g);
  }

  // ---- Encode mode-3 + ReLU ----
  {
    int npix = 128 * 128 * 128;
    encode_mode3_relu<<<(npix + 255) / 256, 256, 0, stream>>>(Y2, P3e, Y3, npix);
  }

  // ---- Decode mode-1: Z1[b,H,wd] = sum_h P1d[h,H] * Y3[b,h,wd] ----
  {
    GemmDesc g;
    g.A = P1d; g.B = Y3; g.C = Z1;
    g.K = 128; g.z0_count = 1;
    g.a_z0 = 0; g.a_z1 = 0;
    g.b_z0 = 0; g.b_z1 = 128L * 256;
    g.c_z0 = 0; g.c_z1 = 256L * 256;
    g.a_ms = 1;   g.a_ks = 256;      // A[m][k] = P1d[k,m]
    g.b_ks = 256;
    g.c_ms = 256; g.c_ns = 1;
    wmma_gemm_f32<<<dim3(256 / TN, 256 / TM, 128), 128, 0, stream>>>(g);
  }

  // ---- Decode mode-2: Z2[b,H,W,d] = sum_w Z1[b,H,w,d] * P2d[w,W] ----
  {
    GemmDesc g;
    g.A = Z1; g.B = P2d; g.C = Z2;
    g.K = 128; g.z0_count = 2;       // z1 = b, z0 = d
    g.a_z0 = 1; g.a_z1 = 256L * 128 * 2;
    g.b_z0 = 0; g.b_z1 = 0;
    g.c_z0 = 1; g.c_z1 = 256L * 256 * 2;
    g.a_ms = 256; g.a_ks = 2;        // Z1[b,m,w,d]
    g.b_ks = 256;
    g.c_ms = 512; g.c_ns = 2;        // Z2[b,m,n,d]
    wmma_gemm_f32<<<dim3(256 / TN, 256 / TM, 256), 128, 0, stream>>>(g);
  }

  // ---- Decode mode-3 ----
  {
    int npix = 128 * 256 * 256;
    decode_mode3<<<(npix + 255) / 256, 256, 0, stream>>>(Z2, P3d, out, npix);
  }
}